// YoloLayer_70781061038443
// MI455X (gfx1250) — compile-verified
//
#include <hip/hip_runtime.h>
#include <math.h>

// ---------------- problem constants (from reference) ----------------
#define NB_   16
#define NA_   3
#define NH_   64
#define NW_   64
#define NT_   50
#define NC_   80
#define CHS_  4096            // nH*nW channel stride
#define CAD_  (NB_*NA_*NH_*NW_)   // 196608
#define BLOCKS_ 768           // CAD_/256
// anchors / STRIDE
__device__ __constant__ float c_aw[3] = {1.875f, 3.875f, 3.6875f};
__device__ __constant__ float c_ah[3] = {3.8125f, 2.8125f, 7.4375f};

// workspace layout (floats)
#define WS_GX 0
#define WS_GY 800
#define WS_GW 1600
#define WS_GH 2400
#define WS_TX 3200
#define WS_TY 4000
#define WS_TW 4800
#define WS_TH 5600
#define WS_TC 6400
#define WS_FL 7200   // int flags: bit0 valid, bit1 winner, bits[15:2] cell key
#define WS_PART 8000 // 768 block partials

typedef float v2f __attribute__((ext_vector_type(2)));
typedef float v8f __attribute__((ext_vector_type(8)));

// ---- wave32 sum reduction via V_WMMA_F32_16X16X4_F32 with B = ones ----
// A[m,0]=v(lane m), A[m,2]=v(lane m+16), A[m,1]=A[m,3]=0  (ISA 16x4 f32 layout)
// D[m,n] = v(m) + v(m+16)  (replicated across columns)
// lane-local sum of the 8 D VGPRs = rows 0..7 (lanes 0-15) or 8..15 (lanes 16-31)
// total = s + shfl_xor(s, 16)
__device__ __forceinline__ float wave_reduce_sum(float v) {
#if __has_builtin(__builtin_amdgcn_wmma_f32_16x16x4_f32)
  v2f A; A[0] = v;    A[1] = 0.0f;
  v2f B; B[0] = 1.0f; B[1] = 1.0f;
  v8f C = {0.f,0.f,0.f,0.f,0.f,0.f,0.f,0.f};
  v8f D = __builtin_amdgcn_wmma_f32_16x16x4_f32(
      /*neg_a=*/false, A, /*neg_b=*/false, B,
      /*c_mod=*/(short)0, C, /*reuse_a=*/false, /*reuse_b=*/false);
  float s = D[0]+D[1]+D[2]+D[3]+D[4]+D[5]+D[6]+D[7];
  s += __shfl_xor(s, 16, 32);
  return s;
#else
  for (int off = 16; off > 0; off >>= 1) v += __shfl_xor(v, off, 32);
  return v;
#endif
}

__device__ __forceinline__ float sigmoidf_(float x) {
  return 1.0f / (1.0f + expf(-x));
}

// ---------------- phase 1: per-target prep (16 blocks x 64 thr) ----------------
__global__ void yolo_prep(const float* __restrict__ target, float* __restrict__ ws) {
  const int b = blockIdx.x;
  const int t = threadIdx.x;
  __shared__ float s_cx[NT_];
  __shared__ int   s_key[NT_];

  if (t < NT_) s_cx[t] = target[b*NT_*5 + t*5 + 1];
  __syncthreads();

  bool valid = false;
  if (t < NT_) {
    valid = true;
    for (int q = 0; q <= t; ++q) valid = valid && (s_cx[q] != 0.0f);

    const int base = b*NT_*5 + t*5;
    float tc = target[base + 0];
    float gx = target[base + 1] * (float)NW_;
    float gy = target[base + 2] * (float)NH_;
    float gw = target[base + 3] * (float)NW_;
    float gh = target[base + 4] * (float)NH_;

    // best anchor by centered w/h IoU (first max wins, as jnp.argmax)
    int best = 0; float bi = -1.0f;
    for (int a = 0; a < 3; ++a) {
      float inter = fminf(gw, c_aw[a]) * fminf(gh, c_ah[a]);
      float uni   = gw*gh + c_aw[a]*c_ah[a] - inter;
      float iou   = inter / uni;
      if (iou > bi) { bi = iou; best = a; }
    }
    int gi = min(max((int)floorf(gx), 0), NW_-1);
    int gj = min(max((int)floorf(gy), 0), NH_-1);
    int key = (best << 12) | (gj << 6) | gi;
    s_key[t] = valid ? key : -1;

    const int o = b*NT_ + t;
    ws[WS_GX + o] = gx;  ws[WS_GY + o] = gy;
    ws[WS_GW + o] = gw;  ws[WS_GH + o] = gh;
    ws[WS_TX + o] = gx - (float)gi;
    ws[WS_TY + o] = gy - (float)gj;
    ws[WS_TW + o] = logf(gw / c_aw[best]);
    ws[WS_TH + o] = logf(gh / c_ah[best]);
    ws[WS_TC + o] = tc;
  }
  __syncthreads();

  if (t < NT_) {
    // last valid t targeting a cell wins (matches the sequential lax.scan .set)
    bool winner = valid;
    if (valid) {
      for (int q = t + 1; q < NT_; ++q)
        if (s_key[q] == s_key[t]) { winner = false; break; }
    }
    int key = valid ? s_key[t] : 0;
    ((int*)ws)[WS_FL + b*NT_ + t] = (valid ? 1 : 0) | (winner ? 2 : 0) | (key << 2);
  }
}

// ---------------- phase 2: per-cell pass (768 blocks x 256 thr) ----------------
__global__ void __launch_bounds__(256)
yolo_main(const float* __restrict__ out, float* __restrict__ ws) {
  __shared__ float s_gx[NT_], s_gy[NT_], s_gw[NT_], s_gh[NT_];
  __shared__ float s_tx[NT_], s_ty[NT_], s_tw[NT_], s_th[NT_], s_tc[NT_];
  __shared__ int   s_fl[NT_];
  __shared__ float s_wave[8];

  const int tid = threadIdx.x;
  const int f   = blockIdx.x * 256 + tid;       // flat b-major cell index
  const int b   = f / (NA_*CHS_);               // one batch per block (256 | 12288)

  if (tid < NT_) {
    const int o = b*NT_ + tid;
    s_gx[tid] = ws[WS_GX + o];  s_gy[tid] = ws[WS_GY + o];
    s_gw[tid] = ws[WS_GW + o];  s_gh[tid] = ws[WS_GH + o];
    s_tx[tid] = ws[WS_TX + o];  s_ty[tid] = ws[WS_TY + o];
    s_tw[tid] = ws[WS_TW + o];  s_th[tid] = ws[WS_TH + o];
    s_tc[tid] = ws[WS_TC + o];
    s_fl[tid] = ((const int*)ws)[WS_FL + o];
  }
  __syncthreads();

  const int r   = f - b*(NA_*CHS_);
  const int a   = r >> 12;          // channel-anchor index
  const int j   = (r >> 6) & 63;
  const int i   = r & 63;
  const int anc = f >> 16;          // reference's b-major read of an a-major anchor array

  const float* cell = out + ((size_t)(b*255 + a*85)) * CHS_ + (j << 6) + i;
  const float r0 = cell[0];
  const float r1 = cell[1*CHS_];
  const float r2 = cell[2*CHS_];
  const float r3 = cell[3*CHS_];
  const float r4 = cell[4*CHS_];

  const float x    = sigmoidf_(r0);
  const float y    = sigmoidf_(r1);
  const float conf = sigmoidf_(r4);
  const float px = x + (float)i;
  const float py = y + (float)j;
  const float pw = expf(r2) * c_aw[anc];
  const float ph = expf(r3) * c_ah[anc];

  const int mykey = (a << 12) | (j << 6) | i;
  float maxiou = 0.0f;
  int found = 0, tci = 0;
  float tx = 0.f, ty = 0.f, tw = 0.f, th = 0.f;

  for (int t = 0; t < NT_; ++t) {
    const int fl = s_fl[t];
    if (!(fl & 1)) break;           // validity is a prefix; uniform per block
    const float gx = s_gx[t], gy = s_gy[t], gw = s_gw[t], gh = s_gh[t];
    const float mx = fminf(px - pw*0.5f, gx - gw*0.5f);
    const float Mx = fmaxf(px + pw*0.5f, gx + gw*0.5f);
    const float my = fminf(py - ph*0.5f, gy - gh*0.5f);
    const float My = fmaxf(py + ph*0.5f, gy + gh*0.5f);
    const float cw = pw + gw - (Mx - mx);
    const float chh = ph + gh - (My - my);
    const float ca = (cw <= 0.f || chh <= 0.f) ? 0.f : cw*chh;
    const float ua = pw*ph + gw*gh - ca;
    maxiou = fmaxf(maxiou, ca / ua);
    if ((fl & 2) && ((fl >> 2) == mykey)) {
      found = 1;
      tx = s_tx[t]; ty = s_ty[t]; tw = s_tw[t]; th = s_th[t];
      tci = (int)s_tc[t];
    }
  }

  float cmask = (maxiou > 0.5f) ? 0.0f : 1.0f;
  float tconf = 0.0f;
  if (found) { cmask = 1.0f; tconf = 1.0f; }
  const float dc = conf - tconf;
  float partial = cmask * dc * dc;                       // loss_conf term

  if (found) {
    const float dx = x - tx, dy = y - ty, dw = r2 - tw, dh = r3 - th;
    partial += 0.5f * (dx*dx + dy*dy + dw*dw + dh*dh);   // loss_coord term
    // online log-softmax over the 80 class logits (strided CHS_ reads,
    // only ~800 of 196608 threads take this path -> ~256 KB total)
    const float* cl = cell + 5*CHS_;
    float m = -INFINITY, ssum = 0.0f, picked = 0.0f;
    for (int c = 0; c < NC_; ++c) {
      const float v = cl[c*CHS_];
      if (c == tci) picked = v;
      if (v > m) { ssum = ssum * expf(m - v); m = v; }
      ssum += expf(v - m);
    }
    partial += -(picked - m - logf(ssum));               // loss_cls term
  }

  // deterministic block reduction: WMMA wave sum + serial 8-way combine
  const float wsum = wave_reduce_sum(partial);
  const int lane = tid & 31, wav = tid >> 5;
  if (lane == 0) s_wave[wav] = wsum;
  __syncthreads();
  if (tid == 0) {
    float acc = 0.f;
    for (int w = 0; w < 8; ++w) acc += s_wave[w];
    ws[WS_PART + blockIdx.x] = acc;
  }
}

// ---------------- phase 3: final reduction (1 block x 32 thr) ----------------
__global__ void yolo_final(const float* __restrict__ ws, float* __restrict__ out) {
  float acc = 0.0f;
  for (int k = threadIdx.x; k < BLOCKS_; k += 32) acc += ws[WS_PART + k];
  const float tot = wave_reduce_sum(acc);
  if (threadIdx.x == 0) out[0] = tot;
}

extern "C" void kernel_launch(void* const* d_in, const int* in_sizes, int n_in,
                              void* d_out, int out_size, void* d_ws, size_t ws_size,
                              hipStream_t stream) {
  (void)in_sizes; (void)n_in; (void)out_size; (void)ws_size;
  const float* output = (const float*)d_in[0];
  const float* target = (const float*)d_in[1];
  float* W = (float*)d_ws;
  float* o = (float*)d_out;

  yolo_prep <<<NB_,     64, 0, stream>>>(target, W);
  yolo_main <<<BLOCKS_, 256, 0, stream>>>(output, W);
  yolo_final<<<1,       32, 0, stream>>>(W, o);
}